// TAMDec_3d_26749056320311
// MI455X (gfx1250) — compile-verified
//
#include <hip/hip_runtime.h>

typedef float v2f __attribute__((ext_vector_type(2)));
typedef float v8f __attribute__((ext_vector_type(8)));

#define B_   4
#define C_   128
#define T_   64
#define HW_  1024   // H*W
#define NSL_ 4      // energy accumulator slices (atomic-contention relief)

// ---------------------------------------------------------------------------
// Kernel 0: zero the sliced energy accumulators (d_ws is poisoned once and
// never re-poisoned; split-K uses atomics, so re-zero on every launch).
// ---------------------------------------------------------------------------
__global__ __launch_bounds__(256) void zero_energy(float* __restrict__ e) {
    int i = blockIdx.x * 256 + threadIdx.x;
    if (i < NSL_ * B_ * T_ * T_) e[i] = 0.0f;
}

// ---------------------------------------------------------------------------
// Kernel 1: energy[b,t,s] += sum_{hw,c} x[b,c,t,hw] * y[b,c,s,hw]
// Split-K over c, TWO c-slabs per block (grid = B*C/2 = 256 blocks, 2048
// waves). 8 waves per block: each wave owns one s-tile and two t-tiles of
// the 64x64 output, reduced with V_WMMA_F32_16X16X4_F32.
//
// f32 16x16x4 fragment layout (ISA 7.12.2): A 16x4 -> lanes 0-15 hold rows,
// VGPR pair holds K={0,1} (lanes 0-15) / K={2,3} (lanes 16-31) => one aligned
// b64 load per lane at column k0 + 2*(lane>>4). B 4x16 mirrored by N-rows.
//
// Partial sums land in slice (blockIdx & 3) of E[NSL][B][64][64]: 1M atomics
// over 64K addresses -> ~16 serialized RMWs per address (vs 128 before).
// ---------------------------------------------------------------------------
__global__ __launch_bounds__(256) void energy_kernel(const float* __restrict__ x,
                                                     const float* __restrict__ y,
                                                     float* __restrict__ E) {
    const int blk = blockIdx.x;
    const int b   = blk / (C_ / 2);
    const int cc  = blk % (C_ / 2);
    const int c0  = cc * 2;

    const int tid  = threadIdx.x;
    const int w    = tid >> 5;
    const int lane = tid & 31;
    const int half = lane >> 4;   // K-pair select
    const int l15  = lane & 15;   // row within tile

    const int s0  = (w & 3) << 4;        // s-tile: 0,16,32,48
    const int t0a = (w >> 2) << 4;       // t-tiles: {0,32} or {16,48}
    const int t0b = t0a + 32;

    v8f acc0 = {};
    v8f acc1 = {};

    for (int sl = 0; sl < 2; ++sl) {
        const float* xb = x + (size_t)(b * C_ + c0 + sl) * T_ * HW_;
        const float* yb = y + (size_t)(b * C_ + c0 + sl) * T_ * HW_;
        const float* arow0 = xb + (size_t)(t0a + l15) * HW_;
        const float* arow1 = xb + (size_t)(t0b + l15) * HW_;
        const float* brow  = yb + (size_t)(s0  + l15) * HW_;

#pragma unroll 4
        for (int k0 = 0; k0 < HW_; k0 += 4) {
            const int off = k0 + 2 * half;
            v2f bf = *(const v2f*)(brow  + off);
            v2f a0 = *(const v2f*)(arow0 + off);
            v2f a1 = *(const v2f*)(arow1 + off);
            acc0 = __builtin_amdgcn_wmma_f32_16x16x4_f32(false, a0, false, bf,
                                                         (short)0, acc0, false, false);
            acc1 = __builtin_amdgcn_wmma_f32_16x16x4_f32(false, a1, false, bf,
                                                         (short)0, acc1, false, false);
        }
    }

    // C/D layout: VGPR r, lane L -> (m = r + 8*(L>>4), n = L&15)
    float* Eb = E + ((size_t)(blk & (NSL_ - 1)) * B_ + b) * T_ * T_;
#pragma unroll
    for (int r = 0; r < 8; ++r) {
        atomicAdd(&Eb[(size_t)(t0a + r + 8 * half) * T_ + s0 + l15], acc0[r]);
        atomicAdd(&Eb[(size_t)(t0b + r + 8 * half) * T_ + s0 + l15], acc1[r]);
    }
}

// ---------------------------------------------------------------------------
// Kernel 2: attention = softmax(rowmax - energy) over last dim, summing the
// NSL accumulator slices on read.
// softmax(rowmax - e) == softmax(-e); stabilized as exp(rowmin - e)/sum.
// One batch per block, one row per wave iteration, wave32 shuffles.
// ---------------------------------------------------------------------------
__global__ __launch_bounds__(256) void softmax_kernel(const float* __restrict__ E,
                                                      float* __restrict__ A) {
    const int b    = blockIdx.x;
    const int tid  = threadIdx.x;
    const int w    = tid >> 5;
    const int lane = tid & 31;

    for (int r = 0; r < 8; ++r) {
        const int t = w * 8 + r;
        const size_t rowoff = ((size_t)b * T_ + t) * T_;

        float e0 = 0.0f, e1 = 0.0f;
#pragma unroll
        for (int sl = 0; sl < NSL_; ++sl) {
            const float* row = E + (size_t)sl * B_ * T_ * T_ + rowoff;
            e0 += row[lane];
            e1 += row[lane + 32];
        }

        float mn = fminf(e0, e1);
#pragma unroll
        for (int off = 16; off > 0; off >>= 1)
            mn = fminf(mn, __shfl_xor(mn, off, 32));
        float p0 = __expf(mn - e0);
        float p1 = __expf(mn - e1);
        float s  = p0 + p1;
#pragma unroll
        for (int off = 16; off > 0; off >>= 1)
            s += __shfl_xor(s, off, 32);
        const float inv = 1.0f / s;

        float* arow = A + rowoff;
        arow[lane]      = p0 * inv;
        arow[lane + 32] = p1 * inv;
    }
}

// ---------------------------------------------------------------------------
// Kernel 3: out[b,c,t,hw] = x[b,c,t,hw] + scale * sum_s att[b,t,s]*y[b,c,s,hw]
// One block per (b,c). Attention tile (64x64, 16 KB) staged in LDS; A-frags
// come from LDS (ds_load_b64), B-frags streamed from y (coalesced b32 pairs,
// each y element read once per block); epilogue fuses the scaled residual.
// ---------------------------------------------------------------------------
__global__ __launch_bounds__(256) void out_kernel(const float* __restrict__ x,
                                                  const float* __restrict__ y,
                                                  const float* __restrict__ scale,
                                                  const float* __restrict__ A,
                                                  float* __restrict__ out) {
    __shared__ float att[T_ * T_];

    const int blk = blockIdx.x;
    const int b   = blk / C_;
    const int c   = blk % C_;
    const float* xb = x   + (size_t)(b * C_ + c) * T_ * HW_;
    const float* yb = y   + (size_t)(b * C_ + c) * T_ * HW_;
    float*       ob = out + (size_t)(b * C_ + c) * T_ * HW_;

    const int tid = threadIdx.x;
    for (int i = tid; i < T_ * T_; i += 256)
        att[i] = A[(size_t)b * T_ * T_ + i];
    __syncthreads();

    const float sc = scale[0];
    const int w    = tid >> 5;
    const int lane = tid & 31;
    const int half = lane >> 4;
    const int l15  = lane & 15;

    for (int nt = w; nt < 64; nt += 8) {
        const int col = nt * 16 + l15;
        v8f acc[4] = {v8f{}, v8f{}, v8f{}, v8f{}};

#pragma unroll 4
        for (int k0 = 0; k0 < T_; k0 += 4) {
            const int s0 = k0 + 2 * half;
            v2f bf;
            bf.x = yb[(size_t)s0 * HW_ + col];
            bf.y = yb[(size_t)(s0 + 1) * HW_ + col];
#pragma unroll
            for (int mt = 0; mt < 4; ++mt) {
                v2f a = *(const v2f*)&att[(mt * 16 + l15) * T_ + s0];
                acc[mt] = __builtin_amdgcn_wmma_f32_16x16x4_f32(false, a, false, bf,
                                                                (short)0, acc[mt],
                                                                false, false);
            }
        }

#pragma unroll
        for (int mt = 0; mt < 4; ++mt) {
#pragma unroll
            for (int r = 0; r < 8; ++r) {
                const int row  = mt * 16 + r + 8 * half;
                const size_t i = (size_t)row * HW_ + col;
                ob[i] = xb[i] + sc * acc[mt][r];
            }
        }
    }
}

// ---------------------------------------------------------------------------
extern "C" void kernel_launch(void* const* d_in, const int* in_sizes, int n_in,
                              void* d_out, int out_size, void* d_ws, size_t ws_size,
                              hipStream_t stream) {
    const float* x     = (const float*)d_in[0];
    const float* y     = (const float*)d_in[1];
    const float* scale = (const float*)d_in[2];
    float* out = (float*)d_out;

    float* E = (float*)d_ws;                  // NSL*B*T*T f32 = 256 KB (sliced energy)
    float* A = E + NSL_ * B_ * T_ * T_;       // B*T*T f32 = 64 KB (attention)

    zero_energy  <<<(NSL_ * B_ * T_ * T_ + 255) / 256, 256, 0, stream>>>(E);
    energy_kernel<<<B_ * C_ / 2, 256, 0, stream>>>(x, y, E);
    softmax_kernel<<<B_, 256, 0, stream>>>(E, A);
    out_kernel   <<<B_ * C_, 256, 0, stream>>>(x, y, scale, A, out);
}